// GCNLayer_7971459301458
// MI455X (gfx1250) — compile-verified
//
#include <hip/hip_runtime.h>

#define N_NODES 100000
#define N_EDGES 1000000
#define DIM 64
#define NTILES (N_NODES / 16)   // 6250, exact

typedef __attribute__((ext_vector_type(2))) float v2f;
typedef __attribute__((ext_vector_type(8))) float v8f;

// ---------------- kernel 1: agg = x (residual folded in), deg = 0 ----------
__global__ __launch_bounds__(256) void gcn_init(const float* __restrict__ x,
                                                float* __restrict__ agg,
                                                unsigned* __restrict__ deg) {
  long long j = (long long)blockIdx.x * blockDim.x + threadIdx.x;
  if (j < (long long)N_NODES * DIM) agg[j] = x[j];
  if (j < N_NODES) deg[j] = 0u;
}

// ---------------- kernel 2: degree histogram (u32 atomics, L2-resident) ----
__global__ __launch_bounds__(256) void gcn_degree(const long long* __restrict__ ei,
                                                  unsigned* __restrict__ deg) {
  int e = blockIdx.x * blockDim.x + threadIdx.x;
  if (e < N_EDGES) {
    long long t = ei[N_EDGES + e];           // target row of edge_index
    atomicAdd(&deg[t], 1u);
  }
}

// ---------------- kernel 3: deg^-1/2 ---------------------------------------
__global__ __launch_bounds__(256) void gcn_dinv(const unsigned* __restrict__ deg,
                                                float* __restrict__ dinv) {
  int i = blockIdx.x * blockDim.x + threadIdx.x;
  if (i < N_NODES) {
    float d = fmaxf((float)deg[i], 1.0f);
    dinv[i] = rsqrtf(d);
  }
}

// ------- kernel 4: per-edge normalized scatter-add (16 lanes / edge) -------
// Each lane: one float4 gather (256B contiguous per edge) + 4 f32 atomics.
__global__ __launch_bounds__(256) void gcn_scatter(const float* __restrict__ x,
                                                   const long long* __restrict__ ei,
                                                   const float* __restrict__ dinv,
                                                   float* __restrict__ agg) {
  long long tid = (long long)blockIdx.x * blockDim.x + threadIdx.x;
  int e = (int)(tid >> 4);
  int j = ((int)tid & 15) * 4;
  if (e >= N_EDGES) return;
  long long s = ei[e];
  long long t = ei[N_EDGES + e];
  float norm = dinv[s] * dinv[t];
  const float4 v = *(const float4*)(x + s * DIM + j);
  float* dst = agg + t * DIM + j;
  atomicAdd(dst + 0, v.x * norm);
  atomicAdd(dst + 1, v.y * norm);
  atomicAdd(dst + 2, v.z * norm);
  atomicAdd(dst + 3, v.w * norm);
}

// ------- kernel 5: out = agg @ W^T + b via V_WMMA_F32_16X16X4_F32 ----------
// One wave per 16-row tile. A: 16x4 f32 fragment = v2f per lane
//   (lanes 0-15: row=lane, K={k0,k0+1}; lanes 16-31: row=lane-16, K={k0+2,k0+3}).
// B[k][n] = W[n][k] -> lane n reads W[n0+n][k0+2h .. +1].
// C/D 16x16 f32: VGPR j, lane l -> row = j + 8*(l>>4), col = l&15.
__global__ __launch_bounds__(256) void gcn_gemm(const float* __restrict__ agg,
                                                const float* __restrict__ W,
                                                const float* __restrict__ bias,
                                                float* __restrict__ out) {
  int lane = threadIdx.x & 31;
  int wave = threadIdx.x >> 5;
  int tile = blockIdx.x * 8 + wave;      // wave-uniform
  if (tile >= NTILES) return;            // uniform exit: EXEC all-ones inside
  int row0 = tile * 16;
  int r = lane & 15;
  int h = lane >> 4;                     // 0 or 1 -> K sub-pair select

  v8f c0 = {}, c1 = {}, c2 = {}, c3 = {};

  const float* arow = agg + (long long)(row0 + r) * DIM + 2 * h;
  const float* w0 = W + (long long)(0 * 16 + r) * DIM + 2 * h;
  const float* w1 = W + (long long)(1 * 16 + r) * DIM + 2 * h;
  const float* w2 = W + (long long)(2 * 16 + r) * DIM + 2 * h;
  const float* w3 = W + (long long)(3 * 16 + r) * DIM + 2 * h;

#pragma unroll
  for (int ks = 0; ks < 16; ++ks) {
    int k0 = ks * 4;
    v2f a;  a.x = arow[k0];  a.y = arow[k0 + 1];
    v2f b0; b0.x = w0[k0];   b0.y = w0[k0 + 1];
    v2f b1; b1.x = w1[k0];   b1.y = w1[k0 + 1];
    v2f b2; b2.x = w2[k0];   b2.y = w2[k0 + 1];
    v2f b3; b3.x = w3[k0];   b3.y = w3[k0 + 1];
    c0 = __builtin_amdgcn_wmma_f32_16x16x4_f32(false, a, false, b0, (short)0, c0, false, false);
    c1 = __builtin_amdgcn_wmma_f32_16x16x4_f32(false, a, false, b1, (short)0, c1, false, false);
    c2 = __builtin_amdgcn_wmma_f32_16x16x4_f32(false, a, false, b2, (short)0, c2, false, false);
    c3 = __builtin_amdgcn_wmma_f32_16x16x4_f32(false, a, false, b3, (short)0, c3, false, false);
  }

  float bb0 = bias[0 * 16 + r];
  float bb1 = bias[1 * 16 + r];
  float bb2 = bias[2 * 16 + r];
  float bb3 = bias[3 * 16 + r];
#pragma unroll
  for (int j = 0; j < 8; ++j) {
    long long orow = (long long)(row0 + j + 8 * h) * DIM;
    out[orow + 0 * 16 + r] = c0[j] + bb0;
    out[orow + 1 * 16 + r] = c1[j] + bb1;
    out[orow + 2 * 16 + r] = c2[j] + bb2;
    out[orow + 3 * 16 + r] = c3[j] + bb3;
  }
}

extern "C" void kernel_launch(void* const* d_in, const int* in_sizes, int n_in,
                              void* d_out, int out_size, void* d_ws, size_t ws_size,
                              hipStream_t stream) {
  const float*     x  = (const float*)d_in[0];
  const long long* ei = (const long long*)d_in[1];   // int64 [2, E]
  const float*     W  = (const float*)d_in[2];       // [64, 64]
  const float*     b  = (const float*)d_in[3];       // [64]
  float*           out = (float*)d_out;

  // workspace layout
  char* ws = (char*)d_ws;
  float*    agg  = (float*)(ws);                                   // N*64 f32 = 25.6 MB
  unsigned* deg  = (unsigned*)(ws + (size_t)N_NODES * DIM * 4);    // N u32   = 0.4 MB
  float*    dinv = (float*)(ws + (size_t)N_NODES * DIM * 4 + (size_t)N_NODES * 4);

  {
    long long total = (long long)N_NODES * DIM;
    int blocks = (int)((total + 255) / 256);
    gcn_init<<<blocks, 256, 0, stream>>>(x, agg, deg);
  }
  gcn_degree<<<(N_EDGES + 255) / 256, 256, 0, stream>>>(ei, deg);
  gcn_dinv<<<(N_NODES + 255) / 256, 256, 0, stream>>>(deg, dinv);
  {
    long long total = (long long)N_EDGES * 16;
    int blocks = (int)((total + 255) / 256);
    gcn_scatter<<<blocks, 256, 0, stream>>>(x, ei, dinv, agg);
  }
  gcn_gemm<<<(NTILES + 7) / 8, 256, 0, stream>>>(agg, W, b, out);
}